// NodePositionLearner_3934190043624
// MI455X (gfx1250) — compile-verified
//
#include <hip/hip_runtime.h>
#include <hip/hip_bf16.h>

#define NN 8192
#define DD 128
#define KK 16
#define K1 17           // K+1: reference keeps top-17 then drops self (rank 0)
#define EPSF 1e-6f
#define MERGE_FLOATS 1088  // per-wave merge area: 16 rows * 2 half-lists * 17 * (d2,idx)

typedef float v2f __attribute__((ext_vector_type(2)));
typedef float v4f __attribute__((ext_vector_type(4)));
typedef float v8f __attribute__((ext_vector_type(8)));

// ---------------- K0: row squared norms + init global min cell ----------------
__global__ void sq_kernel(const float* __restrict__ pos, float* __restrict__ sqg,
                          unsigned* __restrict__ minCell) {
  int i = blockIdx.x * blockDim.x + threadIdx.x;
  if (i == 0) *minCell = 0x7F800000u;  // +inf bits; re-init every launch
  if (i < NN) {
    const v4f* p = (const v4f*)(pos + (size_t)i * DD);
    float s = 0.f;
    #pragma unroll
    for (int t = 0; t < DD / 4; ++t) {
      v4f q = p[t];
      s += q[0] * q[0] + q[1] * q[1] + q[2] * q[2] + q[3] * q[3];
    }
    sqg[i] = s;
  }
}

// ---------------- K1: fused WMMA distances + in-register top-17 ----------------
// Block = 128 threads = 4 independent waves; each wave owns 16 rows.
// Operand roles: A/M side = 16 candidate columns (streamed, exact float2 per
// lane per k-step), B/N side = the wave's own 16 rows (preloaded registers).
// D layout: lane (m,hi), VGPR v -> candidate M = v + 8*hi, own row N = m, so
// every lane accumulates its own row's candidate dot products in registers and
// feeds the sorted top-17 insertion directly — no LDS in the hot loop.
__launch_bounds__(128)
__global__ void knn_kernel(const float* __restrict__ pos, const float* __restrict__ sqg,
                           float* __restrict__ wsd, int* __restrict__ wsi,
                           unsigned* __restrict__ minCell) {
  __shared__ float mlds[4][MERGE_FLOATS];
  const int lane = threadIdx.x & 31;
  const int wave = threadIdx.x >> 5;
  const int hi   = lane >> 4;   // half-wave id
  const int m    = lane & 15;
  const int R    = (blockIdx.x * 4 + wave) * 16;
  float* wbase = mlds[wave];

  // Preload B operand (own rows): lane m carries row R+m; VGPR v at step s -> K = 4s + 2v + hi.
  v2f breg[32];
  {
    const float* rowp = pos + (size_t)(R + m) * DD;
    #pragma unroll
    for (int s = 0; s < 32; ++s) {
      v4f q = *(const v4f*)(rowp + 4 * s);
      v2f b; b[0] = hi ? q[1] : q[0]; b[1] = hi ? q[3] : q[2];
      breg[s] = b;
    }
  }
  const float sqn = sqg[R + m];
  const int rowid = R + m;

  float bd[K1]; int bi[K1];   // sorted ascending (d2, idx); lane covers j = cb + 8*hi + v
  #pragma unroll
  for (int p = 0; p < K1; ++p) { bd[p] = 3.4e38f; bi[p] = -1; }

  for (int cb = 0; cb < NN; cb += 16) {
    // A operand: lane m streams candidate column cb+m; K = 4s + v + 2*hi -> contiguous float2.
    const float* colp = pos + (size_t)(cb + m) * DD + 2 * hi;
    v8f acc = {0.f, 0.f, 0.f, 0.f, 0.f, 0.f, 0.f, 0.f};
    #pragma unroll
    for (int s = 0; s < 32; ++s) {
      v2f a = *(const v2f*)(colp + 4 * s);
      acc = __builtin_amdgcn_wmma_f32_16x16x4_f32(
          false, a, false, breg[s], (short)0, acc, false, false);
    }
    // This lane's 8 candidates: j = cb + 8*hi + v ; sq[j] vectorized (32B aligned).
    v4f sj0 = *(const v4f*)(sqg + cb + 8 * hi);
    v4f sj1 = *(const v4f*)(sqg + cb + 8 * hi + 4);
    #pragma unroll
    for (int v = 0; v < 8; ++v) {
      int j = cb + 8 * hi + v;
      float sqj = (v < 4) ? sj0[v & 3] : sj1[v & 3];
      float d2v = (sqj + sqn) - 2.0f * acc[v];
      if (j == rowid) d2v = -3.4e38f;  // self -> always rank 0, dropped (ref [:,1:])
      if (d2v < bd[K1 - 1]) {
        float cd = d2v; int ci = j;
        #pragma unroll
        for (int p = 0; p < K1; ++p) {  // branch-free sorted insert (fully unrolled)
          bool lt = cd < bd[p];
          float od = bd[p]; int oi = bi[p];
          bd[p] = lt ? cd : od; bi[p] = lt ? ci : oi;
          cd = lt ? od : cd;    ci = lt ? oi : ci;
        }
      }
    }
  }

  // ---- merge the two half-lists per row via (tiny) LDS ----
  {
    int slot = (m * 2 + hi) * K1;
    #pragma unroll
    for (int p = 0; p < K1; ++p) {
      wbase[slot + p] = bd[p];
      wbase[544 + slot + p] = __int_as_float(bi[p]);
    }
  }
  __syncthreads();
  if (hi == 0) {
    const float* Ad = wbase + (m * 2 + 0) * K1;
    const float* Bd = wbase + (m * 2 + 1) * K1;
    const float* Ai = wbase + 544 + (m * 2 + 0) * K1;
    const float* Bi = wbase + 544 + (m * 2 + 1) * K1;
    int ia = 0, ib = 0;
    #pragma unroll 1
    for (int rank = 0; rank < K1; ++rank) {
      float da = (ia < K1) ? Ad[ia] : 3.4e38f;
      float db = (ib < K1) ? Bd[ib] : 3.4e38f;
      int xa = (ia < K1) ? __float_as_int(Ai[ia]) : 0x7fffffff;
      int xb = (ib < K1) ? __float_as_int(Bi[ib]) : 0x7fffffff;
      bool takeA = (da < db) || (da == db && xa < xb);  // jax top_k tie-break: lower idx
      float d2 = takeA ? da : db;
      int  idx = takeA ? xa : xb;
      if (takeA) ++ia; else ++ib;
      if (rank >= 1) {
        float dist = sqrtf(fmaxf(d2, 0.f));
        wsd[(size_t)rowid * KK + (rank - 1)] = dist;
        wsi[(size_t)rowid * KK + (rank - 1)] = idx;
        if (rank == 1)  // nearest kept neighbor -> candidate for global max weight
          atomicMin(minCell, __float_as_uint(dist));  // nonneg floats order as uints
      }
    }
  }
}

// ---------------- K2: batch expansion + weight normalization ----------------
__global__ void expand_kernel(const float* __restrict__ wsd, const int* __restrict__ wsi,
                              const unsigned* __restrict__ minCell,
                              float* __restrict__ out, int B) {
  const long long E = (long long)NN * KK;
  const long long total = (long long)B * E;
  long long e = (long long)blockIdx.x * blockDim.x + threadIdx.x;
  if (e >= total) return;
  int b = (int)(e / E);
  int r = (int)(e % E);
  int n = r / KK;  // source node
  float mind = __uint_as_float(*minCell);
  // weight / max(weight) = (min_d + eps) / (d + eps)
  float w = (mind + EPSF) / (wsd[r] + EPSF);
  long long BE = total;
  out[(long long)b * E + r]          = (float)(n + b * NN);        // edge_index row 0
  out[BE + (long long)b * E + r]     = (float)(wsi[r] + b * NN);   // edge_index row 1
  out[2 * BE + (long long)b * E + r] = w;                          // edge_weight
}

// ---------------- host launcher ----------------
extern "C" void kernel_launch(void* const* d_in, const int* in_sizes, int n_in,
                              void* d_out, int out_size, void* d_ws, size_t ws_size,
                              hipStream_t stream) {
  const float* pos = (const float*)d_in[0];
  // ws layout: [0,4): min-dist bits | 16: sq[N] | +N*4: wsd[N*K] | +N*K*4: wsi[N*K]
  unsigned* minCell = (unsigned*)d_ws;
  float* sqg = (float*)((char*)d_ws + 16);
  float* wsd = (float*)((char*)d_ws + 16 + (size_t)NN * sizeof(float));
  int*   wsi = (int*)((char*)d_ws + 16 + (size_t)NN * sizeof(float)
                      + (size_t)NN * KK * sizeof(float));
  float* out = (float*)d_out;
  const long long E = (long long)NN * KK;
  int B = (int)((long long)out_size / (3 * E));  // recover batch_size from out_size
  if (B < 1) B = 1;

  sq_kernel<<<NN / 256, 256, 0, stream>>>(pos, sqg, minCell);
  knn_kernel<<<NN / 16 / 4, 128, 0, stream>>>(pos, sqg, wsd, wsi, minCell);
  long long total = (long long)B * E;
  expand_kernel<<<(int)((total + 255) / 256), 256, 0, stream>>>(wsd, wsi, minCell, out, B);
}